// PrunableQwenMoeSparseMoeBlockWrapper_69526930587814
// MI455X (gfx1250) — compile-verified
//
#include <hip/hip_runtime.h>
#include <cstddef>

// ---------------------------------------------------------------------------
// MoE block for MI455X (gfx1250, wave32). fp32 numerics => V_WMMA_F32_16X16X4_F32.
// Both GEMM kernels stage tiles with TENSOR_LOAD_TO_LDS (6-arg builtin) into
// ping-pong LDS buffers, overlapped with WMMA compute via S_WAIT_TENSORCNT.
//
// d_out layout: final [1024][2048] f32, then router_logits [1024][8] f32.
// workspace:   cw[1024][8] | sgate[1024] | h_moe[8][1024][1408] | h_sh[1024][5632]
// ---------------------------------------------------------------------------

typedef __attribute__((ext_vector_type(2))) float v2f;
typedef __attribute__((ext_vector_type(8))) float v8f;
typedef unsigned int u32;
typedef unsigned long long u64;
typedef __attribute__((ext_vector_type(4))) u32 v4u;
typedef __attribute__((ext_vector_type(8))) int v8i;
typedef __attribute__((ext_vector_type(4))) int v4i;

static __device__ __forceinline__ v8f wmma_f32(v2f a, v2f b, v8f c) {
  return __builtin_amdgcn_wmma_f32_16x16x4_f32(false, a, false, b, (short)0, c,
                                               false, false);
}

static __device__ __forceinline__ float silu(float x) {
  return x / (1.0f + __expf(-x));
}

// ---------------------------------------------------------------------------
// TDM 2D tile load: global (row-major, stride_d0 elements per row) -> LDS.
// data_size = 4 bytes. Descriptor packing per CDNA5 ISA §8.3/§8.4.
// Optional LDS padding: pad_amount+1 DWORDs inserted every 2^(pad_interval+1)
// DWORDs stored (gives the A tiles a bank-conflict-free row stride of 34).
// ---------------------------------------------------------------------------
static __device__ __forceinline__ void tdm_load_2d(
    u32 lds_byte, const void* gptr, u32 tile_d0, u32 tile_d1, u32 tensor_d0,
    u32 tensor_d1, u32 stride_d0, u32 pad_interval, u32 pad_amount,
    bool pad_en) {
  const u64 ga = (u64)gptr;
  v4u g0;
  g0[0] = 1u;               // count=1, is_restore=0, gather off
  g0[1] = lds_byte;         // lds_addr (bytes)
  g0[2] = (u32)ga;          // global_addr[31:0]
  g0[3] = (u32)(ga >> 32) | (2u << 30);  // global_addr[56:32] | type=2

  v8i g1;
  g1[0] = (int)((2u << 16)                       // data_size = 4B
                | (pad_en ? (1u << 20) : 0u)     // pad_enable
                | (pad_interval << 22)           // pad_interval
                | (pad_amount << 25));           // pad_amount
  g1[1] = (int)((tensor_d0 & 0xFFFFu) << 16);    // tensor_dim0[15:0]
  g1[2] = (int)((tensor_d0 >> 16) | ((tensor_d1 & 0xFFFFu) << 16));
  g1[3] = (int)((tensor_d1 >> 16) | (tile_d0 << 16));  // tile_dim0
  g1[4] = (int)(tile_d1 & 0xFFFFu);              // tile_dim1, tile_dim2=0
  g1[5] = (int)stride_d0;                        // tensor_dim0_stride[31:0]
  g1[6] = 0;                                     // stride hi, dim1_stride lo
  g1[7] = 0;

  const v4i z4 = {0, 0, 0, 0};          // groups 2/3 unused (2D tile)
  const v8i z8 = {0, 0, 0, 0, 0, 0, 0, 0};
  __builtin_amdgcn_tensor_load_to_lds(g0, g1, z4, z4, z8, 0);
}

// ---------------------------------------------------------------------------
// Router: one wave per token. 8 expert logits + shared-expert-gate dot,
// xor-shuffle reduction, softmax + top-2 on lane 0.
// ---------------------------------------------------------------------------
__global__ __launch_bounds__(256) void moe_router_kernel(
    const float* __restrict__ X, const float* __restrict__ Gw,
    const float* __restrict__ Seg, float* __restrict__ cw,
    float* __restrict__ sgate, float* __restrict__ logits_out, int T,
    int Hdim) {
  const int lane = threadIdx.x & 31;
  const int wave = threadIdx.x >> 5;
  const int t = blockIdx.x * 8 + wave;
  if (t >= T) return;

  float l[8] = {0.f, 0.f, 0.f, 0.f, 0.f, 0.f, 0.f, 0.f};
  float accS = 0.f;
  for (int h = lane; h < Hdim; h += 32) {
    const float xv = X[(size_t)t * Hdim + h];
    const float4 g0 = *(const float4*)&Gw[h * 8 + 0];
    const float4 g1 = *(const float4*)&Gw[h * 8 + 4];
    l[0] += xv * g0.x; l[1] += xv * g0.y; l[2] += xv * g0.z; l[3] += xv * g0.w;
    l[4] += xv * g1.x; l[5] += xv * g1.y; l[6] += xv * g1.z; l[7] += xv * g1.w;
    accS += xv * Seg[h];
  }
  #pragma unroll
  for (int off = 16; off > 0; off >>= 1) {
    #pragma unroll
    for (int e = 0; e < 8; ++e) l[e] += __shfl_xor(l[e], off, 32);
    accS += __shfl_xor(accS, off, 32);
  }

  if (lane == 0) {
    float m = l[0];
    #pragma unroll
    for (int e = 1; e < 8; ++e) m = fmaxf(m, l[e]);
    float p[8], s = 0.f;
    #pragma unroll
    for (int e = 0; e < 8; ++e) { p[e] = __expf(l[e] - m); s += p[e]; }
    const float inv = 1.0f / s;
    int i1 = 0;
    #pragma unroll
    for (int e = 1; e < 8; ++e) if (p[e] > p[i1]) i1 = e;
    int i2 = (i1 == 0) ? 1 : 0;
    #pragma unroll
    for (int e = 0; e < 8; ++e)
      if (e != i1 && e != i2 && p[e] > p[i2]) i2 = e;
    #pragma unroll
    for (int e = 0; e < 8; ++e) {
      const float w = (e == i1) ? p[i1] * inv : ((e == i2) ? p[i2] * inv : 0.f);
      cw[(size_t)t * 8 + e] = w;
      logits_out[(size_t)t * 8 + e] = l[e];
    }
    sgate[t] = 1.0f / (1.0f + __expf(-accS));
  }
}

// ---------------------------------------------------------------------------
// Fused gate/up GEMM + silu + per-token scale, TDM-staged + double-buffered.
// Block: 128 threads (4 waves). Tile: 16 tokens x 128 I-cols, K-step 32.
// LDS per buffer: A 16x34 (TDM pad) | Bg 32x128 | Bu 32x128.
// ---------------------------------------------------------------------------
#define GU_A_FLT (16 * 34)
#define GU_B_FLT (32 * 128)
#define GU_BUF_FLT (GU_A_FLT + 2 * GU_B_FLT)

__global__ __launch_bounds__(128) void moe_gateup_silu_kernel(
    const float* __restrict__ X,      // [T][Hdim]
    const float* __restrict__ Wg,     // [E][Hdim][Idim]
    const float* __restrict__ Wu,     // [E][Hdim][Idim]
    float* __restrict__ Hout,         // [E][T][Idim]
    const float* __restrict__ scale,  // [T][sstride]
    int T, int Hdim, int Idim, int sstride) {
  __shared__ float smem[2 * GU_BUF_FLT];
  __shared__ float sScale[16];

  const int tid = threadIdx.x;
  const int lane = tid & 31, wave = tid >> 5;
  const int half = lane >> 4, lr = lane & 15;
  const int i0 = blockIdx.x * 128;
  const int tok0 = blockIdx.y * 16;
  const int e = blockIdx.z;
  const int nb = wave * 32;

  const float* __restrict__ Wge = Wg + (size_t)e * Hdim * Idim;
  const float* __restrict__ Wue = Wu + (size_t)e * Hdim * Idim;

  if (tid < 16) sScale[tid] = scale[(size_t)(tok0 + tid) * sstride + e];

  const u32 ldsBase = (u32)(size_t)(void*)&smem[0];
  const int nk = Hdim / 32;

  // Prologue: stage K-tile 0 into buffer 0.
  if (wave == 0) {
    tdm_load_2d(ldsBase, &X[(size_t)tok0 * Hdim], 32, 16, (u32)Hdim, (u32)T,
                (u32)Hdim, /*pad: 2 DW per 32 DW -> stride 34*/ 4, 1, true);
    tdm_load_2d(ldsBase + GU_A_FLT * 4, &Wge[i0], 128, 32, (u32)Idim,
                (u32)Hdim, (u32)Idim, 0, 0, false);
    tdm_load_2d(ldsBase + (GU_A_FLT + GU_B_FLT) * 4, &Wue[i0], 128, 32,
                (u32)Idim, (u32)Hdim, (u32)Idim, 0, 0, false);
  }

  v8f aG0 = {}, aG1 = {}, aU0 = {}, aU1 = {};

  for (int it = 0; it < nk; ++it) {
    const int cb = (it & 1) ? GU_BUF_FLT : 0;
    const u32 nbB = (it & 1) ? 0u : (u32)(GU_BUF_FLT * 4);

    __syncthreads();  // previous readers of the next-fill buffer are done
    if (wave == 0) {
      if (it + 1 < nk) {
        const int k1 = (it + 1) * 32;
        tdm_load_2d(ldsBase + nbB, &X[(size_t)tok0 * Hdim + k1], 32, 16,
                    (u32)Hdim, (u32)T, (u32)Hdim, 4, 1, true);
        tdm_load_2d(ldsBase + nbB + GU_A_FLT * 4, &Wge[(size_t)k1 * Idim + i0],
                    128, 32, (u32)Idim, (u32)Hdim, (u32)Idim, 0, 0, false);
        tdm_load_2d(ldsBase + nbB + (GU_A_FLT + GU_B_FLT) * 4,
                    &Wue[(size_t)k1 * Idim + i0], 128, 32, (u32)Idim,
                    (u32)Hdim, (u32)Idim, 0, 0, false);
        __builtin_amdgcn_s_wait_tensorcnt(3);  // current tile's 3 loads done
      } else {
        __builtin_amdgcn_s_wait_tensorcnt(0);
      }
    }
    __syncthreads();  // current buffer published to all waves

    const float* __restrict__ Ab = &smem[cb];
    const float* __restrict__ Bg = &smem[cb + GU_A_FLT];
    const float* __restrict__ Bu = &smem[cb + GU_A_FLT + GU_B_FLT];
    const int c0 = nb + lr, c1 = nb + 16 + lr;

    #pragma unroll
    for (int kk = 0; kk < 32; kk += 4) {
      const int ko = kk + half * 2;  // lanes 16-31 take K+2,K+3
      const v2f a = *(const v2f*)&Ab[lr * 34 + ko];
      const v2f bg0 = {Bg[(ko + 0) * 128 + c0], Bg[(ko + 1) * 128 + c0]};
      const v2f bg1 = {Bg[(ko + 0) * 128 + c1], Bg[(ko + 1) * 128 + c1]};
      const v2f bu0 = {Bu[(ko + 0) * 128 + c0], Bu[(ko + 1) * 128 + c0]};
      const v2f bu1 = {Bu[(ko + 0) * 128 + c1], Bu[(ko + 1) * 128 + c1]};
      aG0 = wmma_f32(a, bg0, aG0);
      aG1 = wmma_f32(a, bg1, aG1);
      aU0 = wmma_f32(a, bu0, aU0);
      aU1 = wmma_f32(a, bu1, aU1);
    }
  }

  float* __restrict__ out = Hout + (size_t)e * T * Idim;
  #pragma unroll
  for (int j = 0; j < 8; ++j) {
    const int row = j + half * 8;  // C layout: VGPR j = rows j / j+8
    const float sc = sScale[row];
    const size_t rb = (size_t)(tok0 + row) * Idim + i0 + nb;
    const float g0 = aG0[j], u0 = aU0[j];
    out[rb + lr] = silu(g0) * u0 * sc;
    const float g1 = aG1[j], u1 = aU1[j];
    out[rb + 16 + lr] = silu(g1) * u1 * sc;
  }
}

// ---------------------------------------------------------------------------
// Down-projection + combine, TDM-staged + double-buffered across a flattened
// K of 8*Idim (experts) + ISdim (shared; sigmoid gate pre-folded into h_sh):
// 528 K-tiles of 32, pipelined across segment boundaries.
// ---------------------------------------------------------------------------
#define DN_A_FLT (16 * 34)
#define DN_B_FLT (32 * 128)
#define DN_BUF_FLT (DN_A_FLT + DN_B_FLT)

__global__ __launch_bounds__(128) void moe_down_combine_kernel(
    const float* __restrict__ Hmoe,  // [E][T][Idim]
    const float* __restrict__ Wd,    // [E][Idim][Hdim]
    const float* __restrict__ Hsh,   // [T][ISdim]
    const float* __restrict__ Wsd,   // [ISdim][Hdim]
    float* __restrict__ Out,         // [T][Hdim]
    int T, int Hdim, int Idim, int ISdim, int E) {
  __shared__ float smem[2 * DN_BUF_FLT];

  const int tid = threadIdx.x;
  const int lane = tid & 31, wave = tid >> 5;
  const int half = lane >> 4, lr = lane & 15;
  const int n0 = blockIdx.x * 128;
  const int tok0 = blockIdx.y * 16;
  const int nb = wave * 32;

  const u32 ldsBase = (u32)(size_t)(void*)&smem[0];
  const int tilesPerExp = Idim / 32;            // 44
  const int moeTiles = E * tilesPerExp;         // 352
  const int nTiles = moeTiles + ISdim / 32;     // 528

  // flat K-tile index -> A tile ptr / A row stride / B tile ptr
  auto tile_src = [&](int idx, const float*& A, u32& lda, const float*& B) {
    if (idx < moeTiles) {
      const int seg = idx / tilesPerExp;
      const int k0 = (idx - seg * tilesPerExp) * 32;
      A = Hmoe + ((size_t)seg * T + tok0) * Idim + k0;
      lda = (u32)Idim;
      B = Wd + (size_t)seg * Idim * Hdim + (size_t)k0 * Hdim + n0;
    } else {
      const int k0 = (idx - moeTiles) * 32;
      A = Hsh + (size_t)tok0 * ISdim + k0;
      lda = (u32)ISdim;
      B = Wsd + (size_t)k0 * Hdim + n0;
    }
  };

  // Prologue: stage flat tile 0 into buffer 0.
  if (wave == 0) {
    const float* A; const float* B; u32 lda;
    tile_src(0, A, lda, B);
    tdm_load_2d(ldsBase, A, 32, 16, lda, (u32)T, lda, 4, 1, true);
    tdm_load_2d(ldsBase + DN_A_FLT * 4, B, 128, 32, (u32)Hdim, 32u, (u32)Hdim,
                0, 0, false);
  }

  v8f c0 = {}, c1 = {};

  for (int it = 0; it < nTiles; ++it) {
    const int cb = (it & 1) ? DN_BUF_FLT : 0;
    const u32 nbB = (it & 1) ? 0u : (u32)(DN_BUF_FLT * 4);

    __syncthreads();
    if (wave == 0) {
      if (it + 1 < nTiles) {
        const float* A; const float* B; u32 lda;
        tile_src(it + 1, A, lda, B);
        tdm_load_2d(ldsBase + nbB, A, 32, 16, lda, (u32)T, lda, 4, 1, true);
        tdm_load_2d(ldsBase + nbB + DN_A_FLT * 4, B, 128, 32, (u32)Hdim, 32u,
                    (u32)Hdim, 0, 0, false);
        __builtin_amdgcn_s_wait_tensorcnt(2);  // current tile's 2 loads done
      } else {
        __builtin_amdgcn_s_wait_tensorcnt(0);
      }
    }
    __syncthreads();

    const float* __restrict__ Ab = &smem[cb];
    const float* __restrict__ Bt = &smem[cb + DN_A_FLT];
    const int cc0 = nb + lr, cc1 = nb + 16 + lr;

    #pragma unroll
    for (int kk = 0; kk < 32; kk += 4) {
      const int ko = kk + half * 2;
      const v2f a = *(const v2f*)&Ab[lr * 34 + ko];
      const v2f b0 = {Bt[(ko + 0) * 128 + cc0], Bt[(ko + 1) * 128 + cc0]};
      const v2f b1 = {Bt[(ko + 0) * 128 + cc1], Bt[(ko + 1) * 128 + cc1]};
      c0 = wmma_f32(a, b0, c0);
      c1 = wmma_f32(a, b1, c1);
    }
  }

  #pragma unroll
  for (int j = 0; j < 8; ++j) {
    const int row = j + half * 8;
    const size_t rb = (size_t)(tok0 + row) * Hdim + n0 + nb;
    Out[rb + lr] = c0[j];
    Out[rb + 16 + lr] = c1[j];
  }
}

// ---------------------------------------------------------------------------
extern "C" void kernel_launch(void* const* d_in, const int* in_sizes, int n_in,
                              void* d_out, int out_size, void* d_ws,
                              size_t ws_size, hipStream_t stream) {
  const float* X   = (const float*)d_in[0];  // [2,512,2048]
  const float* Gw  = (const float*)d_in[1];  // [2048,8]
  const float* Wg  = (const float*)d_in[2];  // [8,2048,1408]
  const float* Wu  = (const float*)d_in[3];  // [8,2048,1408]
  const float* Wd  = (const float*)d_in[4];  // [8,1408,2048]
  const float* Sg  = (const float*)d_in[5];  // [2048,5632]
  const float* Su  = (const float*)d_in[6];  // [2048,5632]
  const float* Sd  = (const float*)d_in[7];  // [5632,2048]
  const float* Seg = (const float*)d_in[8];  // [2048,1]

  const int T = 1024, H = 2048, E = 8, I = 1408, IS = 5632;

  float* out    = (float*)d_out;            // final [T][H]
  float* logits = out + (size_t)T * H;      // router_logits [T][8]

  float* cw    = (float*)d_ws;              // [T][8]
  float* sgate = cw + (size_t)T * E;        // [T]
  float* h_moe = sgate + T;                 // [E][T][I]
  float* h_sh  = h_moe + (size_t)E * T * I; // [T][IS]

  moe_router_kernel<<<T / 8, 256, 0, stream>>>(X, Gw, Seg, cw, sgate, logits,
                                               T, H);

  dim3 g1(I / 128, T / 16, E);
  moe_gateup_silu_kernel<<<g1, 128, 0, stream>>>(X, Wg, Wu, h_moe, cw, T, H, I,
                                                 E);

  dim3 g2(IS / 128, T / 16, 1);
  moe_gateup_silu_kernel<<<g2, 128, 0, stream>>>(X, Sg, Su, h_sh, sgate, T, H,
                                                 IS, 1);

  dim3 g3(H / 128, T / 16, 1);
  moe_down_combine_kernel<<<g3, 128, 0, stream>>>(h_moe, Wd, h_sh, Sd, out, T,
                                                  H, I, IS, E);
}